// DFMSDA_36034775614174
// MI455X (gfx1250) — compile-verified
//
#include <hip/hip_runtime.h>

typedef float v2f __attribute__((ext_vector_type(2)));
typedef float v8f __attribute__((ext_vector_type(8)));

#define DIMC     256
#define HEAD_DIM 32
#define NDIL     4
#define DPG      64
#define NB       8
#define NH       64
#define NW       64
#define NPIX     4096            // NH*NW
#define NPIX_TOT 32768           // NB*NPIX
#define QKV_ROWS 768
#define ATT_SCALE 0.17677669529663687f  // 32^-0.5
#define LN_EPS 1e-5f

// ---------------------------------------------------------------------------
// K1: sub = vi - ir ; xn = LayerNorm_c(sub)  (both stored [b, c, n])
// ---------------------------------------------------------------------------
__global__ void k_sub_ln(const float* __restrict__ vi, const float* __restrict__ ir,
                         const float* __restrict__ ln_g, const float* __restrict__ ln_b,
                         float* __restrict__ sub, float* __restrict__ xn) {
  int p = blockIdx.x * blockDim.x + threadIdx.x;      // pixel id, 0..32767
  int b = p >> 12, n = p & 4095;
  size_t base = (size_t)b * DIMC * NPIX + n;
  float s0 = 0.f, s1 = 0.f;
  for (int c = 0; c < DIMC; ++c) {
    size_t i = base + (size_t)c * NPIX;
    float s = vi[i] - ir[i];
    sub[i] = s;
    s0 += s; s1 += s * s;
  }
  float mu = s0 * (1.0f / DIMC);
  float var = s1 * (1.0f / DIMC) - mu * mu;
  float rstd = __frsqrt_rn(var + LN_EPS);
  for (int c = 0; c < DIMC; ++c) {
    size_t i = base + (size_t)c * NPIX;
    xn[i] = (sub[i] - mu) * rstd * ln_g[c] + ln_b[c];
  }
}

// ---------------------------------------------------------------------------
// K2: qkv[b,o,n] = sum_c w_qkv[o,c] * xn[b,c,n]   via V_WMMA_F32_16X16X4_F32
// one wave = one 16(M) x 32(N) pair of tiles sharing the A fragment
// ---------------------------------------------------------------------------
__global__ void k_qkv_gemm(const float* __restrict__ w, const float* __restrict__ xn,
                           float* __restrict__ qkv) {
  int lane  = threadIdx.x & 31;
  int wave  = threadIdx.x >> 5;
  int pt    = blockIdx.x * 8 + wave;     // 0 .. 49151  (pair-tile id)
  int b     = pt / 6144;                 // 48 * 128 pair-tiles per batch
  int r     = pt % 6144;
  int mo    = (r >> 7) << 4;             // output-row tile (0..767)
  int no    = (r & 127) << 5;            // pixel pair-tile (0..4095, width 32)
  int row16 = lane & 15;
  int khalf = (lane >> 4) << 1;          // 0 or 2

  const float* xb = xn + (size_t)b * DIMC * NPIX;
  v8f acc0 = {}, acc1 = {};
  for (int k0 = 0; k0 < DIMC; k0 += 4) {
    v2f a, b0, b1;
    a.x  = w[(mo + row16) * DIMC + k0 + khalf];
    a.y  = w[(mo + row16) * DIMC + k0 + khalf + 1];
    size_t r0 = (size_t)(k0 + khalf) * NPIX;
    b0.x = xb[r0 + no + row16];
    b0.y = xb[r0 + NPIX + no + row16];
    b1.x = xb[r0 + no + 16 + row16];
    b1.y = xb[r0 + NPIX + no + 16 + row16];
    acc0 = __builtin_amdgcn_wmma_f32_16x16x4_f32(false, a, false, b0,
                                                 (short)0, acc0, false, false);
    acc1 = __builtin_amdgcn_wmma_f32_16x16x4_f32(false, a, false, b1,
                                                 (short)0, acc1, false, false);
  }
  int mhi = (lane >> 4) << 3;  // +8 for upper half-lanes
  float* qb = qkv + (size_t)b * QKV_ROWS * NPIX;
#pragma unroll
  for (int j = 0; j < 8; ++j) {
    size_t row = (size_t)(mo + j + mhi) * NPIX;
    qb[row + no + row16]      = acc0[j];
    qb[row + no + 16 + row16] = acc1[j];
  }
}

// ---------------------------------------------------------------------------
// K3: multi-dilation 3x3 local attention (9 taps, softmax), VALU kernel.
// one thread = (b, group, head, pixel). ybuf layout [p, ch] with
// ch = g*64 + h*32 + c  (matches reference channel order)
// ---------------------------------------------------------------------------
__global__ void k_attention(const float* __restrict__ qkv, float* __restrict__ ybuf) {
  int idx = blockIdx.x * blockDim.x + threadIdx.x;    // 0 .. 262143
  int n    = idx & 4095;
  int rest = idx >> 12;
  int h    = rest & 1;
  int g    = (rest >> 1) & 3;
  int b    = rest >> 3;
  int py   = n >> 6, px = n & 63;
  int dil  = g + 1;
  int cbase = g * DPG + h * HEAD_DIM;

  const float* qb = qkv + ((size_t)b * QKV_ROWS + cbase) * NPIX;          // s=0
  const float* kb = qkv + ((size_t)b * QKV_ROWS + DIMC + cbase) * NPIX;   // s=1
  const float* vb = qkv + ((size_t)b * QKV_ROWS + 2 * DIMC + cbase) * NPIX;

  float q[HEAD_DIM];
#pragma unroll
  for (int c = 0; c < HEAD_DIM; ++c) q[c] = qb[(size_t)c * NPIX + n];

  float lg[9];
#pragma unroll
  for (int t = 0; t < 9; ++t) {
    int ky = py + (t / 3 - 1) * dil;
    int kx = px + (t % 3 - 1) * dil;
    float dot = 0.f;
    if (ky >= 0 && ky < NH && kx >= 0 && kx < NW) {
      int nn = ky * NW + kx;
      for (int c = 0; c < HEAD_DIM; ++c) dot += q[c] * kb[(size_t)c * NPIX + nn];
    }
    lg[t] = dot * ATT_SCALE;   // zero-padded taps contribute logit 0 (as reference)
  }
  float m = lg[0];
#pragma unroll
  for (int t = 1; t < 9; ++t) m = fmaxf(m, lg[t]);
  float z = 0.f;
#pragma unroll
  for (int t = 0; t < 9; ++t) { lg[t] = __expf(lg[t] - m); z += lg[t]; }
  float rz = 1.0f / z;

  float out[HEAD_DIM];
#pragma unroll
  for (int c = 0; c < HEAD_DIM; ++c) out[c] = 0.f;
#pragma unroll
  for (int t = 0; t < 9; ++t) {
    int ky = py + (t / 3 - 1) * dil;
    int kx = px + (t % 3 - 1) * dil;
    if (ky >= 0 && ky < NH && kx >= 0 && kx < NW) {
      int nn = ky * NW + kx;
      float wgt = lg[t] * rz;
      for (int c = 0; c < HEAD_DIM; ++c) out[c] += wgt * vb[(size_t)c * NPIX + nn];
    }
  }
  float* yb = ybuf + ((size_t)b * NPIX + n) * DIMC + cbase;
#pragma unroll
  for (int c = 0; c < HEAD_DIM; ++c) yb[c] = out[c];
}

// ---------------------------------------------------------------------------
// K4: div[b,oc,n] = sub[b,oc,n] + proj_b[oc] + sum_ch ybuf[p,ch]*proj_w[oc,ch]
// WMMA GEMM, M = pixels, N = out channels; 2 N-tiles per wave share A fragment
// ---------------------------------------------------------------------------
__global__ void k_proj_res(const float* __restrict__ ybuf, const float* __restrict__ pw,
                           const float* __restrict__ pb, const float* __restrict__ sub,
                           float* __restrict__ divb) {
  int lane  = threadIdx.x & 31;
  int wave  = threadIdx.x >> 5;
  int t     = blockIdx.x * 8 + wave;     // 0 .. 16383  (pair-tile id)
  int mo    = (t >> 3) << 4;             // pixel tile (2048 of them)
  int no    = (t & 7) << 5;              // out-channel pair-tile (width 32)
  int row16 = lane & 15;
  int khalf = (lane >> 4) << 1;

  v8f acc0 = {}, acc1 = {};
  for (int k0 = 0; k0 < DIMC; k0 += 4) {
    v2f a, b0, b1;
    a.x  = ybuf[(size_t)(mo + row16) * DIMC + k0 + khalf];
    a.y  = ybuf[(size_t)(mo + row16) * DIMC + k0 + khalf + 1];
    b0.x = pw[(size_t)(no + row16) * DIMC + k0 + khalf];       // proj_w[o,ch]
    b0.y = pw[(size_t)(no + row16) * DIMC + k0 + khalf + 1];
    b1.x = pw[(size_t)(no + 16 + row16) * DIMC + k0 + khalf];
    b1.y = pw[(size_t)(no + 16 + row16) * DIMC + k0 + khalf + 1];
    acc0 = __builtin_amdgcn_wmma_f32_16x16x4_f32(false, a, false, b0,
                                                 (short)0, acc0, false, false);
    acc1 = __builtin_amdgcn_wmma_f32_16x16x4_f32(false, a, false, b1,
                                                 (short)0, acc1, false, false);
  }
  int mhi = (lane >> 4) << 3;
  int oc0 = no + row16;
  int oc1 = no + 16 + row16;
  float bias0 = pb[oc0], bias1 = pb[oc1];
#pragma unroll
  for (int j = 0; j < 8; ++j) {
    int p = mo + j + mhi;
    int b = p >> 12, n = p & 4095;
    size_t i0 = ((size_t)b * DIMC + oc0) * NPIX + n;
    size_t i1 = ((size_t)b * DIMC + oc1) * NPIX + n;
    divb[i0] = sub[i0] + acc0[j] + bias0;
    divb[i1] = sub[i1] + acc1[j] + bias1;
  }
}

// ---------------------------------------------------------------------------
// K5: o[b,c] = vi[b,c] (64x64) @ div[b,c] (64x64), 2048 pairs; duplicate out.
// One block per (b,c). Both operands staged in LDS via
// GLOBAL_LOAD_ASYNC_TO_LDS_B128 (ASYNCcnt path). The LDS addresses passed to
// the asm are ptrtoint of the real __shared__ addresses (low 32 bits of the
// generic pointer = LDS byte offset), which also captures smem so the
// "memory" clobber keeps the subsequent ds_load reads alive.
// ---------------------------------------------------------------------------
__global__ void k_final_bmm(const float* __restrict__ vi, const float* __restrict__ divb,
                            float* __restrict__ out) {
  __shared__ __align__(16) float smem[8192];    // [0,4096)=A, [4096,8192)=B
  size_t base = (size_t)blockIdx.x * 4096;
  int tid = threadIdx.x;

  // async stage: each thread pushes 4 x 16B per matrix, lane-coalesced
#pragma unroll
  for (int j = 0; j < 4; ++j) {
    int i = (tid << 2) + (j << 10);             // float index, 16B aligned
    const float* ga = vi + base + i;
    const float* gb = divb + base + i;
    unsigned la = (unsigned)(size_t)(smem + i);         // LDS byte offset of A
    unsigned lb = (unsigned)(size_t)(smem + 4096 + i);  // LDS byte offset of B
    asm volatile("global_load_async_to_lds_b128 %0, %1, off"
                 :: "v"(la), "v"(ga) : "memory");
    asm volatile("global_load_async_to_lds_b128 %0, %1, off"
                 :: "v"(lb), "v"(gb) : "memory");
  }
  asm volatile("s_wait_asynccnt 0x0" ::: "memory");
  __syncthreads();

  const float* As = smem;
  const float* Bs = smem + 4096;
  int lane  = tid & 31;
  int wave  = tid >> 5;
  int row16 = lane & 15;
  int khalf = (lane >> 4) << 1;
  int mhi   = (lane >> 4) << 3;
  int mo    = (wave & 3) << 4;                  // 4 M-tiles
  int no    = (wave >> 2) << 5;                 // 2 N-pair-tiles (width 32)

  v8f acc0 = {}, acc1 = {};
#pragma unroll
  for (int k0 = 0; k0 < 64; k0 += 4) {
    v2f a, b0, b1;
    a.x  = As[(mo + row16) * 64 + k0 + khalf];
    a.y  = As[(mo + row16) * 64 + k0 + khalf + 1];
    b0.x = Bs[(k0 + khalf) * 64 + no + row16];
    b0.y = Bs[(k0 + khalf + 1) * 64 + no + row16];
    b1.x = Bs[(k0 + khalf) * 64 + no + 16 + row16];
    b1.y = Bs[(k0 + khalf + 1) * 64 + no + 16 + row16];
    acc0 = __builtin_amdgcn_wmma_f32_16x16x4_f32(false, a, false, b0,
                                                 (short)0, acc0, false, false);
    acc1 = __builtin_amdgcn_wmma_f32_16x16x4_f32(false, a, false, b1,
                                                 (short)0, acc1, false, false);
  }
#pragma unroll
  for (int j = 0; j < 8; ++j) {
    size_t o0 = base + (size_t)(mo + j + mhi) * 64 + no + row16;
    float v0 = acc0[j], v1 = acc1[j];
    out[o0]      = v0;
    out[o0 + 16] = v1;
    out[o0 + (size_t)2048 * 4096]      = v0;    // tuple duplicate (o, o)
    out[o0 + 16 + (size_t)2048 * 4096] = v1;
  }
}

// ---------------------------------------------------------------------------
extern "C" void kernel_launch(void* const* d_in, const int* in_sizes, int n_in,
                              void* d_out, int out_size, void* d_ws, size_t ws_size,
                              hipStream_t stream) {
  const float* vi     = (const float*)d_in[0];
  const float* ir     = (const float*)d_in[1];
  const float* w_qkv  = (const float*)d_in[2];
  const float* proj_w = (const float*)d_in[3];
  const float* proj_b = (const float*)d_in[4];
  const float* ln_g   = (const float*)d_in[5];
  const float* ln_b   = (const float*)d_in[6];
  float* out = (float*)d_out;

  const size_t PLANE = (size_t)NB * DIMC * NPIX;   // 8.39M floats
  float* sub  = (float*)d_ws;
  float* xn   = sub + PLANE;                        // reused as ybuf after K2
  float* qkv  = xn + PLANE;                         // 25.2M floats; reused as div
  float* ybuf = xn;
  float* divb = qkv;

  // K1: sub + LayerNorm  (32768 pixels)
  k_sub_ln<<<128, 256, 0, stream>>>(vi, ir, ln_g, ln_b, sub, xn);
  // K2: qkv GEMM, 49152 wave pair-tiles, 8 waves/block
  k_qkv_gemm<<<6144, 256, 0, stream>>>(w_qkv, xn, qkv);
  // K3: dilated local attention, 262144 threads
  k_attention<<<1024, 256, 0, stream>>>(qkv, ybuf);
  // K4: proj GEMM + bias + residual, 16384 wave pair-tiles
  k_proj_res<<<2048, 256, 0, stream>>>(ybuf, proj_w, proj_b, sub, divb);
  // K5: 2048 batched 64x64x64 f32 matmuls (async LDS staging), duplicated out
  k_final_bmm<<<2048, 256, 0, stream>>>(vi, divb, out);
}